// ContrastiveLoss_44135083934242
// MI455X (gfx1250) — compile-verified
//
#include <hip/hip_runtime.h>
#include <hip/hip_bf16.h>

#define KCLS 64
#define DDIM 256

typedef float v2f __attribute__((ext_vector_type(2)));
typedef float v8f __attribute__((ext_vector_type(8)));

// ---------------------------------------------------------------------------
// Kernel 1: per-class prototype sums via one-hot GEMM on V_WMMA_F32_16X16X4_F32
//   sums[c, d] += sum_s (label[s]==c) * feat[s, d]
// Block = 256 threads = 8 waves. Wave w owns dims [32w, 32w+32) (2 N-tiles)
// and all 4 class M-tiles -> 8 f32 16x16 accumulators (64 VGPRs).
// Main loop has NO bounds checks (EXEC stays all-1s, no exec save/restore);
// the ragged tail (N%4 != 0) is handled by a separate predicated loop.
// ---------------------------------------------------------------------------
__global__ __launch_bounds__(256)
void proto_wmma_kernel(const float* __restrict__ feat,
                       const int* __restrict__ label,
                       float* __restrict__ sums,
                       int N) {
    const int wave = threadIdx.x >> 5;   // 0..7
    const int lane = threadIdx.x & 31;
    const int m    = lane & 15;          // tile row (class) / tile col (dim)
    const int half = lane >> 4;          // 0 or 1
    const int kk0  = half * 2;           // per-lane K sub-index (A/B VGPR0)
    const int kk1  = kk0 + 1;            // per-lane K sub-index (A/B VGPR1)

    const int groups  = (N + 3) >> 2;    // 4 samples per step
    const int fullG   = N >> 2;          // groups with all 4 samples valid
    const int gPerBlk = (groups + gridDim.x - 1) / gridDim.x;
    const int g0 = blockIdx.x * gPerBlk;
    int g1 = g0 + gPerBlk; if (g1 > groups) g1 = groups;
    int gMid = g1 < fullG ? g1 : fullG;  // end of unconditional region
    if (gMid < g0) gMid = g0;

    const int dBase = wave * 32;
    const float* fBase = feat + dBase + m;   // per-lane column base

    v8f acc[4][2];
    for (int ct = 0; ct < 4; ++ct)
        for (int t = 0; t < 2; ++t)
            for (int e = 0; e < 8; ++e) acc[ct][t][e] = 0.0f;

    // ---- main loop: fully in-bounds, branch-free, EXEC all-1s ----
    for (int g = g0; g < gMid; ++g) {
        const int sBase = g << 2;
        // labels at wave-uniform addresses -> scalar (s_load) path
        const int l0 = label[sBase + 0];
        const int l1 = label[sBase + 1];
        const int l2 = label[sBase + 2];
        const int l3 = label[sBase + 3];
        const int lab0 = half ? l2 : l0;
        const int lab1 = half ? l3 : l1;

        const float* r0 = fBase + (size_t)(sBase + kk0) * DDIM;
        const float* r1 = fBase + (size_t)(sBase + kk1) * DDIM;
        v2f b0, b1;                       // B fragments: rows kk0/kk1, col m
        b0.x = r0[0];
        b0.y = r1[0];
        b1.x = r0[16];
        b1.y = r1[16];

        // stream prefetch ~8 groups (32 rows) ahead; speculative, safe at end
        __builtin_prefetch(r0 + 32 * DDIM, 0, 0);

        for (int ct = 0; ct < 4; ++ct) {  // 4 class tiles cover K=64
            v2f a;                        // one-hot A fragment (exact 0/1 f32)
            a.x = (lab0 == ct * 16 + m) ? 1.0f : 0.0f;
            a.y = (lab1 == ct * 16 + m) ? 1.0f : 0.0f;
            acc[ct][0] = __builtin_amdgcn_wmma_f32_16x16x4_f32(
                false, a, false, b0, (short)0, acc[ct][0], false, false);
            acc[ct][1] = __builtin_amdgcn_wmma_f32_16x16x4_f32(
                false, a, false, b1, (short)0, acc[ct][1], false, false);
        }
    }

    // ---- ragged tail (at most one group in the whole grid) ----
    for (int g = gMid; g < g1; ++g) {
        const int sBase = g << 2;
        const int s0 = sBase + kk0;
        const int s1 = sBase + kk1;
        const bool ok0 = s0 < N;
        const bool ok1 = s1 < N;
        const int cs0 = ok0 ? s0 : (N - 1);
        const int cs1 = ok1 ? s1 : (N - 1);
        const int lab0 = label[cs0];
        const int lab1 = label[cs1];

        const float* r0 = fBase + (size_t)cs0 * DDIM;
        const float* r1 = fBase + (size_t)cs1 * DDIM;
        v2f b0, b1;
        b0.x = ok0 ? r0[0]  : 0.0f;
        b0.y = ok1 ? r1[0]  : 0.0f;
        b1.x = ok0 ? r0[16] : 0.0f;
        b1.y = ok1 ? r1[16] : 0.0f;

        for (int ct = 0; ct < 4; ++ct) {
            v2f a;
            a.x = (ok0 && (lab0 == ct * 16 + m)) ? 1.0f : 0.0f;
            a.y = (ok1 && (lab1 == ct * 16 + m)) ? 1.0f : 0.0f;
            acc[ct][0] = __builtin_amdgcn_wmma_f32_16x16x4_f32(
                false, a, false, b0, (short)0, acc[ct][0], false, false);
            acc[ct][1] = __builtin_amdgcn_wmma_f32_16x16x4_f32(
                false, a, false, b1, (short)0, acc[ct][1], false, false);
        }
    }

    // Flush C tiles: VGPR v, lane -> element (ct*16 + v + 8*half, dBase + t*16 + m)
    for (int ct = 0; ct < 4; ++ct)
        for (int t = 0; t < 2; ++t) {
            const int col = dBase + t * 16 + m;
            for (int v = 0; v < 8; ++v) {
                const int row = ct * 16 + v + half * 8;
                atomicAdd(&sums[row * DDIM + col], acc[ct][t][v]);
            }
        }
}

// ---------------------------------------------------------------------------
// Kernel 2: per-class counts (LDS-privatized histogram)
// ---------------------------------------------------------------------------
__global__ __launch_bounds__(256)
void count_kernel(const int* __restrict__ label, float* __restrict__ cnt, int N) {
    __shared__ unsigned int lc[KCLS];
    for (int i = threadIdx.x; i < KCLS; i += blockDim.x) lc[i] = 0u;
    __syncthreads();
    for (int i = blockIdx.x * blockDim.x + threadIdx.x; i < N;
         i += gridDim.x * blockDim.x)
        atomicAdd(&lc[label[i]], 1u);
    __syncthreads();
    for (int i = threadIdx.x; i < KCLS; i += blockDim.x)
        if (lc[i]) atomicAdd(&cnt[i], (float)lc[i]);
}

// ---------------------------------------------------------------------------
// Kernel 3: prototype = sum/count, then L2-normalize each row
// One wave32 per row; 8 blocks x 8 waves = 64 rows.
// ---------------------------------------------------------------------------
__global__ __launch_bounds__(256)
void normalize_kernel(const float* __restrict__ sums,
                      const float* __restrict__ cnt,
                      float* __restrict__ outp) {
    const int wave = threadIdx.x >> 5;
    const int lane = threadIdx.x & 31;
    const int row  = blockIdx.x * 8 + wave;
    const float inv_c = 1.0f / cnt[row];

    float vals[8];
    float ss = 0.0f;
    for (int i = 0; i < 8; ++i) {
        float v = sums[row * DDIM + lane + i * 32] * inv_c;
        vals[i] = v;
        ss += v * v;
    }
    for (int off = 16; off > 0; off >>= 1)
        ss += __shfl_xor(ss, off, 32);          // wave32 reduction
    const float scale = 1.0f / fmaxf(sqrtf(ss), 1e-12f);
    for (int i = 0; i < 8; ++i)
        outp[row * DDIM + lane + i * 32] = vals[i] * scale;
}

// ---------------------------------------------------------------------------
// Kernel 4: InfoNCE-style loss.
// Block k (64 blocks), thread d (256). log(Fp) = a*r/tau exactly, so
// -log(Fp/(Fn+eps)) = log(Fn+eps) - a*r/tau.
// ---------------------------------------------------------------------------
__global__ __launch_bounds__(256)
void loss_kernel(const float* __restrict__ NA,
                 const float* __restrict__ NR,
                 float* __restrict__ out) {
    const int k = blockIdx.x;
    const int d = threadIdx.x;
    const float invTau = 2.0f;                  // 1 / 0.5
    const float a = NA[k * DDIM + d];
    const float r = NR[k * DDIM + d];
    const float fplog = a * r * invTau;
    const float Fp = __expf(fplog);

    float accum = 0.0f;
    for (int j = 0; j < KCLS; ++j) {
        const float sa = __expf(a * NA[j * DDIM + d] * invTau);
        const float sr = __expf(a * NR[j * DDIM + d] * invTau);
        if (j != k) accum += sa + sr;
    }
    const float Fn = accum + 2.0f * (float)(KCLS - 1) * Fp;
    const float contrib = (__logf(Fn + 1e-8f) - fplog) * (1.0f / (float)DDIM);

    __shared__ float red[256];
    red[threadIdx.x] = contrib;
    __syncthreads();
    for (int s = 128; s > 0; s >>= 1) {
        if (threadIdx.x < s) red[threadIdx.x] += red[threadIdx.x + s];
        __syncthreads();
    }
    if (threadIdx.x == 0) atomicAdd(out, red[0]);
}

// ---------------------------------------------------------------------------
extern "C" void kernel_launch(void* const* d_in, const int* in_sizes, int n_in,
                              void* d_out, int out_size, void* d_ws, size_t ws_size,
                              hipStream_t stream) {
    const float* atacF = (const float*)d_in[0];
    const float* rnaF  = (const float*)d_in[1];
    const int*   atacL = (const int*)d_in[2];
    const int*   rnaL  = (const int*)d_in[3];
    float* out = (float*)d_out;

    const int N = in_sizes[0] / DDIM;

    float* ws   = (float*)d_ws;
    float* sumA = ws;                          // [64*256]
    float* sumR = ws + KCLS * DDIM;            // [64*256]
    float* cntA = ws + 2 * KCLS * DDIM;        // [64]
    float* cntR = cntA + KCLS;                 // [64]
    float* nA   = cntR + KCLS;                 // [64*256]
    float* nR   = nA + KCLS * DDIM;            // [64*256]

    hipMemsetAsync(d_ws, 0, (size_t)(2 * KCLS * DDIM + 2 * KCLS) * sizeof(float),
                   stream);
    hipMemsetAsync(d_out, 0, sizeof(float), stream);

    count_kernel<<<64, 256, 0, stream>>>(atacL, cntA, N);
    count_kernel<<<64, 256, 0, stream>>>(rnaL, cntR, N);

    proto_wmma_kernel<<<304, 256, 0, stream>>>(atacF, atacL, sumA, N);
    proto_wmma_kernel<<<304, 256, 0, stream>>>(rnaF, rnaL, sumR, N);

    normalize_kernel<<<8, 256, 0, stream>>>(sumA, cntA, nA);
    normalize_kernel<<<8, 256, 0, stream>>>(sumR, cntR, nR);

    loss_kernel<<<64, 256, 0, stream>>>(nA, nR, out);
}